// LSTMHardSigmoid_5669356834689
// MI455X (gfx1250) — compile-verified
//
#include <hip/hip_runtime.h>
#include <hip/hip_bf16.h>

// LSTM w/ hard-sigmoid gates on MI455X (gfx1250), fp32 throughout using
// V_WMMA_F32_16X16X4_F32 so the 512-step recurrence keeps reference precision.
//
// Per timestep (one launch, graph-captured):
//   gates[b, g*H+h] = [x_t | h_prev] (64x1024)  @  [w_ih | w_hh].T (1024x2048) + b_ih + b_hh
// Grid: (H/16, B/16) blocks, 128 threads (4 waves). Wave g computes the 16x16
// tile of gate g for this (batch-tile, hidden-tile); gates meet in LDS and the
// block finishes the cell update (c, h, out[t]) itself.
//
// Latency note: with ~1 wave/SIMD there is no cross-wave latency hiding, so
// the WMMA accumulation chain is the critical path. We keep TWO independent
// accumulators (even/odd K-chunks) to halve the exposed WMMA->WMMA RAW depth,
// and sum them once at the end.
//
// Workspace (d_ws): h ping (B*H f32) | h pong (B*H f32) | c (B*H f32) = 384 KB.
// h is double-buffered because every block reads all H columns of h_prev while
// producing its own 16-column slice of h_next.

typedef __attribute__((ext_vector_type(2))) float v2f;
typedef __attribute__((ext_vector_type(8))) float v8f;

#define T_LEN 512
#define B_DIM 64
#define I_DIM 512
#define H_DIM 512

__device__ __forceinline__ float hard_sig(float x) {
    return __builtin_fminf(__builtin_fmaxf(__builtin_fmaf(0.2f, x, 0.5f), 0.0f), 1.0f);
}

__global__ void lstm_init(const float* __restrict__ h0, const float* __restrict__ c0,
                          float* __restrict__ h_ws, float* __restrict__ c_ws) {
    int i = blockIdx.x * blockDim.x + threadIdx.x;
    if (i < B_DIM * H_DIM) {
        h_ws[i] = h0[i];
        c_ws[i] = c0[i];
    }
}

__global__ __launch_bounds__(128)
void lstm_step(const float* __restrict__ x,     // [T, B, I]
               const float* __restrict__ w_ih,  // [4H, I] row-major
               const float* __restrict__ w_hh,  // [4H, H] row-major
               const float* __restrict__ b_ih,  // [4H]
               const float* __restrict__ b_hh,  // [4H]
               const float* __restrict__ h_rd,  // [B, H] h_{t-1}
               float* __restrict__ h_wr,        // [B, H] h_t
               float* __restrict__ c_ws,        // [B, H] in-place cell state
               float* __restrict__ out,         // [T, B, H]
               float* __restrict__ hn,          // [B, H] final h
               float* __restrict__ cn,          // [B, H] final c
               int t, int write_final)
{
    __shared__ float lds[4][256];          // [gate][16x16 tile, VGPR-row major]

    const int lane = threadIdx.x & 31;
    const int gate = threadIdx.x >> 5;     // 0:i 1:f 2:g 3:o  (one wave each)
    const int hb   = blockIdx.x * 16;      // hidden-column tile base
    const int mb   = blockIdx.y * 16;      // batch-row tile base

    // WMMA f32 16x16x4 per-lane fragment addressing (ISA 7.12.2):
    //   A: lane holds row M = lane&15, K-pair base 2*(lane>>4), VGPR v -> K = base+v
    //   B: lane holds col N = lane&15, same K-pair mapping
    const int mrow = lane & 15;
    const int kb   = (lane >> 4) << 1;     // 0 or 2
    const int ng   = gate * H_DIM + hb + (lane & 15);   // global gate column

    const float* xrow  = x + ((size_t)t * B_DIM + (mb + mrow)) * I_DIM + kb;
    const float* hrow  = h_rd + (size_t)(mb + mrow) * H_DIM + kb;
    const float* wirow = w_ih + (size_t)ng * I_DIM + kb;
    const float* whrow = w_hh + (size_t)ng * H_DIM + kb;

    __builtin_prefetch(hrow, 0, 0);        // global_prefetch_b8: warm h row while x-phase runs

    // Fold both biases into accumulator 0 (same column -> same bias for all 8 rows).
    const float bias = b_ih[ng] + b_hh[ng];
    v8f acc0 = {bias, bias, bias, bias, bias, bias, bias, bias};
    v8f acc1 = {0.f, 0.f, 0.f, 0.f, 0.f, 0.f, 0.f, 0.f};

    // --- input contribution: x_t (16xI) @ w_ih.T (Ix16) ---
    #pragma unroll 2
    for (int k = 0; k < I_DIM; k += 8) {
        v2f a0 = *(const v2f*)(xrow  + k);
        v2f b0 = *(const v2f*)(wirow + k);
        v2f a1 = *(const v2f*)(xrow  + k + 4);
        v2f b1 = *(const v2f*)(wirow + k + 4);
        acc0 = __builtin_amdgcn_wmma_f32_16x16x4_f32(false, a0, false, b0,
                                                     (short)0, acc0, false, false);
        acc1 = __builtin_amdgcn_wmma_f32_16x16x4_f32(false, a1, false, b1,
                                                     (short)0, acc1, false, false);
    }
    // --- recurrent contribution: h_{t-1} (16xH) @ w_hh.T (Hx16) ---
    #pragma unroll 2
    for (int k = 0; k < H_DIM; k += 8) {
        v2f a0 = *(const v2f*)(hrow  + k);
        v2f b0 = *(const v2f*)(whrow + k);
        v2f a1 = *(const v2f*)(hrow  + k + 4);
        v2f b1 = *(const v2f*)(whrow + k + 4);
        acc0 = __builtin_amdgcn_wmma_f32_16x16x4_f32(false, a0, false, b0,
                                                     (short)0, acc0, false, false);
        acc1 = __builtin_amdgcn_wmma_f32_16x16x4_f32(false, a1, false, b1,
                                                     (short)0, acc1, false, false);
    }

    const v8f acc = acc0 + acc1;

    // Publish this gate's 16x16 tile. D layout: VGPR r, lane l -> M=r+8*(l>>4), N=l&15.
    #pragma unroll
    for (int r = 0; r < 8; ++r) lds[gate][r * 32 + lane] = acc[r];
    __syncthreads();

    // 256 tile elements, 128 threads -> 2 each. Same storage index s for all 4 gates.
    #pragma unroll
    for (int p = 0; p < 2; ++p) {
        const int s   = threadIdx.x + p * 128;
        const int row = (s >> 5) + ((s >> 4) & 1) * 8;   // M within tile
        const int col = s & 15;                          // N within tile

        const float iv = hard_sig(lds[0][s]);
        const float fv = hard_sig(lds[1][s]);
        const float gv = tanhf(lds[2][s]);
        const float ov = hard_sig(lds[3][s]);

        const int bi = mb + row;
        const int hc = hb + col;
        const size_t idx = (size_t)bi * H_DIM + hc;

        const float c_new = fv * c_ws[idx] + iv * gv;
        const float h_new = ov * tanhf(c_new);

        c_ws[idx] = c_new;
        h_wr[idx] = h_new;
        out[(size_t)t * B_DIM * H_DIM + idx] = h_new;
        if (write_final) {
            hn[idx] = h_new;
            cn[idx] = c_new;
        }
    }
}

extern "C" void kernel_launch(void* const* d_in, const int* in_sizes, int n_in,
                              void* d_out, int out_size, void* d_ws, size_t ws_size,
                              hipStream_t stream) {
    (void)in_sizes; (void)n_in; (void)out_size; (void)ws_size;

    const float* x    = (const float*)d_in[0];
    const float* h0   = (const float*)d_in[1];
    const float* c0   = (const float*)d_in[2];
    const float* w_ih = (const float*)d_in[3];
    const float* w_hh = (const float*)d_in[4];
    const float* b_ih = (const float*)d_in[5];
    const float* b_hh = (const float*)d_in[6];

    float* out = (float*)d_out;                       // [T,B,H]
    float* hn  = out + (size_t)T_LEN * B_DIM * H_DIM; // [B,H]
    float* cn  = hn + (size_t)B_DIM * H_DIM;          // [B,H]

    float* h_a  = (float*)d_ws;
    float* h_b  = h_a + (size_t)B_DIM * H_DIM;
    float* c_ws = h_b + (size_t)B_DIM * H_DIM;

    lstm_init<<<(B_DIM * H_DIM + 255) / 256, 256, 0, stream>>>(h0, c0, h_a, c_ws);

    dim3 grid(H_DIM / 16, B_DIM / 16);   // 32 x 4 blocks
    for (int t = 0; t < T_LEN; ++t) {
        float* hr = (t & 1) ? h_b : h_a;
        float* hw = (t & 1) ? h_a : h_b;
        lstm_step<<<grid, 128, 0, stream>>>(x, w_ih, w_hh, b_ih, b_hh,
                                            hr, hw, c_ws, out, hn, cn,
                                            t, (t == T_LEN - 1) ? 1 : 0);
    }
}